// FocalLoss_24438363914777
// MI455X (gfx1250) — compile-verified
//
#include <hip/hip_runtime.h>
#include <hip/hip_bf16.h>
#include <math.h>

#define NUM_CLASSES 90
#define NA 49104
#define NB 8
#define NM 32
#define AC (NA * NUM_CLASSES)   /* 4419360, divisible by 4 */
#define EPSV 1e-4f
#define REG_BLOCKS 192          /* ceil(49104/256) */
#define CLS_BLOCKS 512

typedef float v2f __attribute__((ext_vector_type(2)));
typedef float v8f __attribute__((ext_vector_type(8)));

// Full-precision wave32 reduction via one f32 WMMA.
// A-matrix layout (32-bit, 16x4): lanes 0-15 -> {K0,K1}, lanes 16-31 -> {K2,K3}.
// Put s in VGPR0, 0 in VGPR1 => A[m][0]=s_m, A[m][2]=s_{m+16}.
// B = all ones => D[m][n] = s_m + s_{m+16}.
// C/D layout: VGPR r holds M=r (lanes 0-15) and M=8+r (lanes 16-31), so the
// lane-local sum of the 8 result VGPRs gives: lanes 0-15 -> sum(s_0..7,s_16..23),
// lanes 16-31 -> sum(s_8..15,s_24..31). Halves are folded in LDS below.
__device__ __forceinline__ float wave_reduce_half_wmma(float s) {
  v2f a; a[0] = s;    a[1] = 0.0f;
  v2f b; b[0] = 1.0f; b[1] = 1.0f;
  v8f c = {0.f, 0.f, 0.f, 0.f, 0.f, 0.f, 0.f, 0.f};
  c = __builtin_amdgcn_wmma_f32_16x16x4_f32(false, a, false, b, (short)0, c,
                                            false, false);
  return c[0] + c[1] + c[2] + c[3] + c[4] + c[5] + c[6] + c[7];
}

// Deterministic block reduction: WMMA per wave, halves to LDS, thread 0 folds
// in fixed order and writes the block partial. All 256 threads must arrive.
__device__ __forceinline__ void block_reduce_store(float val, float* dest,
                                                   float* lds16) {
  float h = wave_reduce_half_wmma(val);
  int lane = threadIdx.x & 31;
  int wave = threadIdx.x >> 5;
  if (lane == 0)  lds16[wave * 2 + 0] = h;
  if (lane == 16) lds16[wave * 2 + 1] = h;
  __syncthreads();
  if (threadIdx.x == 0) {
    float t = 0.0f;
    for (int i = 0; i < 16; ++i) t += lds16[i];
    *dest = t;
  }
}

__global__ void init_kernel(int* npos) {
  if (threadIdx.x < NB) npos[threadIdx.x] = 0;
}

// One thread per anchor: IoU assignment, per-anchor class code, num_pos,
// smooth-L1 regression loss (positives only).
__global__ void assign_kernel(const float* __restrict__ anchors,   // [A][4] y1,x1,y2,x2
                              const float* __restrict__ ann,       // [B][32][5] x1,y1,x2,y2,lbl
                              const float* __restrict__ reg,       // [B][A][4]
                              signed char* __restrict__ codes,     // [B][A]
                              int* __restrict__ npos,              // [B]
                              float* __restrict__ reg_part) {      // [B][REG_BLOCKS]
  __shared__ float sann[NM * 5];
  __shared__ float lds16[16];
  const int b = blockIdx.y;
  if (threadIdx.x < NM * 5) sann[threadIdx.x] = ann[b * NM * 5 + threadIdx.x];
  __syncthreads();

  const int a = blockIdx.x * blockDim.x + threadIdx.x;
  const bool av = (a < NA);
  const int ac = av ? a : (NA - 1);

  const float ay1 = anchors[ac * 4 + 0];
  const float ax1 = anchors[ac * 4 + 1];
  const float ay2 = anchors[ac * 4 + 2];
  const float ax2 = anchors[ac * 4 + 3];
  const float aarea = (ay2 - ay1) * (ax2 - ax1);

  float best = -2.0f;
  int arg = 0;
  for (int m = 0; m < NM; ++m) {
    const float bx1 = sann[m * 5 + 0], by1 = sann[m * 5 + 1];
    const float bx2 = sann[m * 5 + 2], by2 = sann[m * 5 + 3];
    const float lb  = sann[m * 5 + 4];
    float iw = fminf(ax2, bx2) - fmaxf(ax1, bx1);
    float ih = fminf(ay2, by2) - fmaxf(ay1, by1);
    iw = fmaxf(iw, 0.0f);
    ih = fmaxf(ih, 0.0f);
    const float inter = iw * ih;
    const float ua = fmaxf(aarea + (bx2 - bx1) * (by2 - by1) - inter, 1e-8f);
    float iou = inter / ua;
    if (lb == -1.0f) iou = -1.0f;
    if (iou > best) { best = iou; arg = m; }  // first-max == jnp.argmax
  }

  const bool pos = (best >= 0.5f);
  const signed char code =
      pos ? (signed char)(int)sann[arg * 5 + 4]
          : ((best < 0.4f) ? (signed char)-1 : (signed char)-2);
  if (av) codes[(size_t)b * NA + a] = code;
  if (av && pos) atomicAdd(&npos[b], 1);

  float rsum = 0.0f;
  if (av && pos) {
    const float gx1 = sann[arg * 5 + 0], gy1 = sann[arg * 5 + 1];
    const float gx2 = sann[arg * 5 + 2], gy2 = sann[arg * 5 + 3];
    float gw = gx2 - gx1, gh = gy2 - gy1;
    const float gcx = gx1 + 0.5f * gw, gcy = gy1 + 0.5f * gh;
    gw = fmaxf(gw, 1.0f);
    gh = fmaxf(gh, 1.0f);
    const float aw = ax2 - ax1, ah = ay2 - ay1;
    const float acx = ax1 + 0.5f * aw, acy = ay1 + 0.5f * ah;
    float tt[4];
    tt[0] = (gcy - acy) / ah;       // t_dy
    tt[1] = (gcx - acx) / aw;       // t_dx
    tt[2] = logf(gh / ah);          // t_dh
    tt[3] = logf(gw / aw);          // t_dw
    const float* rp = reg + ((size_t)b * NA + a) * 4;
    #pragma unroll
    for (int j = 0; j < 4; ++j) {
      const float d = fabsf(tt[j] - rp[j]);
      rsum += (d <= (1.0f / 9.0f)) ? 4.5f * d * d : d - (0.5f / 9.0f);
    }
  }
  block_reduce_store(rsum, &reg_part[b * REG_BLOCKS + blockIdx.x], lds16);
}

// Streaming focal-BCE over the 141MB classification slab (bandwidth-bound).
__global__ void cls_kernel(const float* __restrict__ cls,
                           const signed char* __restrict__ codes,
                           float* __restrict__ cls_part) {  // [B][CLS_BLOCKS]
  __shared__ float lds16[16];
  const int b = blockIdx.y;
  const float4* cp = (const float4*)(cls + (size_t)b * AC);
  const signed char* cb = codes + (size_t)b * NA;
  const unsigned n4 = AC / 4u;
  const unsigned stride = gridDim.x * blockDim.x;

  float acc = 0.0f;
  for (unsigned i = blockIdx.x * blockDim.x + threadIdx.x; i < n4; i += stride) {
    const float4 v = cp[i];
    __builtin_prefetch(cp + i + stride, 0, 0);  // global_prefetch_b8, speculative
    unsigned e = i * 4u;
    unsigned a = e / 90u;
    unsigned c = e - a * 90u;
    const float vv[4] = {v.x, v.y, v.z, v.w};
    #pragma unroll
    for (int j = 0; j < 4; ++j) {
      const float p = fminf(fmaxf(vv[j], EPSV), 1.0f - EPSV);
      const int code = cb[a];
      float loss;
      if (code == -2) {                 // ignore band (0.4 <= iou < 0.5)
        loss = 0.0f;
      } else if (code == (int)c) {      // positive anchor, target class: t=1
        const float q = 1.0f - p;
        loss = -0.5f * q * q * __logf(p);
      } else {                          // t=0 (negatives + pos off-class)
        loss = -0.5f * p * p * __logf(1.0f - p);
      }
      acc += loss;
      if (++c == 90u) { c = 0u; ++a; }
    }
  }
  block_reduce_store(acc, &cls_part[b * CLS_BLOCKS + blockIdx.x], lds16);
}

__global__ void finalize_kernel(const float* __restrict__ cls_part,
                                const float* __restrict__ reg_part,
                                const int* __restrict__ npos,
                                float* __restrict__ out) {
  __shared__ float cl[NB], rl[NB];
  const int b = threadIdx.x;
  if (b < NB) {
    float cs = 0.0f;
    for (int i = 0; i < CLS_BLOCKS; ++i) cs += cls_part[b * CLS_BLOCKS + i];
    float rs = 0.0f;
    for (int i = 0; i < REG_BLOCKS; ++i) rs += reg_part[b * REG_BLOCKS + i];
    const int np = npos[b];
    const float npf = (float)np;
    cl[b] = cs / fmaxf(npf, 1.0f);
    rl[b] = (np > 0) ? rs / fmaxf(npf * 4.0f, 1.0f) : 0.0f;
  }
  __syncthreads();
  if (threadIdx.x == 0) {
    float c = 0.0f, r = 0.0f;
    for (int i = 0; i < NB; ++i) { c += cl[i]; r += rl[i]; }
    out[0] = c * (1.0f / NB);
    out[1] = r * (1.0f / NB);
  }
}

extern "C" void kernel_launch(void* const* d_in, const int* in_sizes, int n_in,
                              void* d_out, int out_size, void* d_ws, size_t ws_size,
                              hipStream_t stream) {
  (void)in_sizes; (void)n_in; (void)out_size; (void)ws_size;
  const float* cls = (const float*)d_in[0];  // [B,A,C]
  const float* reg = (const float*)d_in[1];  // [B,A,4]
  const float* anc = (const float*)d_in[2];  // [1,A,4]
  const float* ann = (const float*)d_in[3];  // [B,32,5]
  float* out = (float*)d_out;                // [2]

  char* ws = (char*)d_ws;
  signed char* codes = (signed char*)ws;                     // B*A bytes
  size_t off = ((size_t)NB * NA + 511) & ~(size_t)511;
  float* cls_part = (float*)(ws + off);                      // NB*CLS_BLOCKS
  float* reg_part = cls_part + (size_t)NB * CLS_BLOCKS;      // NB*REG_BLOCKS
  int*   npos     = (int*)(reg_part + (size_t)NB * REG_BLOCKS);

  init_kernel<<<1, 32, 0, stream>>>(npos);
  assign_kernel<<<dim3(REG_BLOCKS, NB), 256, 0, stream>>>(anc, ann, reg, codes,
                                                          npos, reg_part);
  cls_kernel<<<dim3(CLS_BLOCKS, NB), 256, 0, stream>>>(cls, codes, cls_part);
  finalize_kernel<<<1, 32, 0, stream>>>(cls_part, reg_part, npos, out);
}